// Decoder_3848290697640
// MI455X (gfx1250) — compile-verified
//
#include <hip/hip_runtime.h>
#include <stdint.h>

typedef float v2f __attribute__((ext_vector_type(2)));
typedef float v8f __attribute__((ext_vector_type(8)));

#define HIDDEN 128

// ---------------------------------------------------------------------------
// Kernel 0: build visited bitmask (idempotent under duplicate indices),
// compute q = [x;X[s];X[p]] @ Wq, then qk = (Wk @ q) / sqrt(128).
// ---------------------------------------------------------------------------
__global__ __launch_bounds__(256) void prep_kernel(
    const float* __restrict__ X, const float* __restrict__ xv,
    const float* __restrict__ Wq, const float* __restrict__ Wk,
    const int* __restrict__ visited, const int* __restrict__ snp,
    const int* __restrict__ pnp,
    float* __restrict__ qk, unsigned* __restrict__ maskW,
    int nWords, int nVisited)
{
    __shared__ float fc[3 * HIDDEN];
    __shared__ float q[HIDDEN];
    const int t = threadIdx.x;

    for (int i = t; i < nWords; i += 256) maskW[i] = 0u;
    __syncthreads();
    for (int i = t; i < nVisited; i += 256) {
        int idx = visited[i];
        atomicOr(&maskW[idx >> 5], 1u << (idx & 31));
    }

    const int sn = snp[0], pn = pnp[0];
    if (t < HIDDEN) {
        fc[t]       = xv[t];
        fc[128 + t] = X[(size_t)sn * HIDDEN + t];
        fc[256 + t] = X[(size_t)pn * HIDDEN + t];
    }
    __syncthreads();
    if (t < HIDDEN) {
        float a = 0.f;
        for (int i = 0; i < 3 * HIDDEN; ++i) a += fc[i] * Wq[i * HIDDEN + t];
        q[t] = a;
    }
    __syncthreads();
    if (t < HIDDEN) {
        float a = 0.f;
        for (int j = 0; j < HIDDEN; ++j) a += Wk[t * HIDDEN + j] * q[j];
        qk[t] = a * 0.08838834764831845f; // 1/sqrt(128)
    }
}

// ---------------------------------------------------------------------------
// Kernel 1: the streaming pass.  Each wave owns a contiguous span of 16-row
// tiles of X. Tiles are async-copied global->LDS (double buffered, ASYNCcnt).
// u^T(1x16) = qk^T(1x128) @ Xtile^T via v_wmma_f32_16x16x4_f32 with
//   A = qk chunk (row M=0; rows 1..15 are don't-care, no masking needed)
//   B = Xtile K-chunk (4x16) straight from LDS
// so u[n] lands in acc[0] of lane n (D: vgpr0, lanes0-15 = M=0,N=lane).
// Online softmax state {m, s, y[128]} is updated per tile.
// ---------------------------------------------------------------------------
__device__ __forceinline__ void issue_tile_async(uint64_t xbase, unsigned tileByte,
                                                 unsigned ldsBase, int lane)
{
    unsigned go0 = tileByte + (unsigned)lane * 16u;  // 16B per lane, 512B per instr
    unsigned lo0 = ldsBase + (unsigned)lane * 16u;
#pragma unroll
    for (int i = 0; i < 16; ++i) {                   // 16 x 512B = 8KB tile
        unsigned lo = lo0 + (unsigned)(i * 512);
        unsigned go = go0 + (unsigned)(i * 512);
        asm volatile("global_load_async_to_lds_b128 %0, %1, %2"
                     :: "v"(lo), "v"(go), "s"(xbase) : "memory");
    }
}

__global__ __launch_bounds__(64) void stream_kernel(
    const float* __restrict__ X, const float* __restrict__ qk,
    const unsigned* __restrict__ maskW, float* __restrict__ partials,
    int span, int tiles)
{
    // dynamic LDS layout (floats): [2 waves][2 bufs][2048] -> 8192, qk -> 128
    extern __shared__ float smem[];
    const int lane = threadIdx.x & 31;
    const int wv   = threadIdx.x >> 5;
    const int gw   = blockIdx.x * 2 + wv;

    float* qkL = smem + 8192;
    for (int i = threadIdx.x; i < HIDDEN; i += 64) qkL[i] = qk[i];
    __syncthreads();

    const unsigned bufByte0 = (unsigned)wv * 16384u;
    const uint64_t xbase = (uint64_t)(uintptr_t)X;

    int t0   = gw * span;
    int tend = t0 + span; if (tend > tiles) tend = tiles;

    float mcur = -INFINITY, scur = 0.f;
    float4 y; y.x = y.y = y.z = y.w = 0.f;

    const bool hi   = lane >= 16;
    const int  brow = hi ? lane - 16 : lane;  // B layout: lanes0-15 K0/1, lanes16-31 K2/3
    const int  koff = hi ? 2 : 0;

    // Hoist A operands (qk chunks) out of the tile loop: loop-invariant,
    // only lanes 0/16 feed the M=0 row that we read back; others don't-care.
    v2f aq[32];
#pragma unroll
    for (int kc = 0; kc < 32; ++kc)
        aq[kc] = *(const v2f*)(qkL + 4 * kc + koff);

    if (t0 < tend) {
        issue_tile_async(xbase, (unsigned)t0 * 8192u, bufByte0, lane);
        for (int tt = t0; tt < tend; ++tt) {
            const int b = (tt - t0) & 1;
            if (tt + 1 < tend) {
                issue_tile_async(xbase, (unsigned)(tt + 1) * 8192u,
                                 bufByte0 + (unsigned)((b ^ 1) * 8192), lane);
                asm volatile("s_wait_asynccnt 16" ::: "memory"); // tile tt complete
            } else {
                asm volatile("s_wait_asynccnt 0" ::: "memory");
            }
            const float* buf = smem + wv * 4096 + b * 2048;

            // ---- u(16) via 32 K-chunks of 4: D row M=0 accumulates qk . X[n,:]
            v8f acc = {};
#pragma unroll
            for (int kc = 0; kc < 32; ++kc) {
                v2f bx = *(const v2f*)(buf + brow * HIDDEN + 4 * kc + koff);
                acc = __builtin_amdgcn_wmma_f32_16x16x4_f32(
                          false, aq[kc], false, bx, (short)0, acc, false, false);
            }

            float um = -INFINITY;
            if (lane < 16) {
                int rowg = tt * 16 + lane;
                unsigned mwd = maskW[rowg >> 5];
                float bias = ((mwd >> (rowg & 31)) & 1u) ? 0.f : 1.f; // visited -> +0
                um = acc[0] + bias;   // D(M=0, N=lane)
            }
            // tile max over lanes 0..15, broadcast
            float mt = um;
            mt = fmaxf(mt, __shfl_xor(mt, 1));
            mt = fmaxf(mt, __shfl_xor(mt, 2));
            mt = fmaxf(mt, __shfl_xor(mt, 4));
            mt = fmaxf(mt, __shfl_xor(mt, 8));
            mt = __shfl(mt, 0);

            const float mnew  = fmaxf(mcur, mt);
            const float scale = __expf(mcur - mnew);   // exp(-inf)=0 on first tile
            const float w = (lane < 16) ? __expf(um - mnew) : 0.f;
            float sw = w;
            sw += __shfl_xor(sw, 16);
            sw += __shfl_xor(sw, 8);
            sw += __shfl_xor(sw, 4);
            sw += __shfl_xor(sw, 2);
            sw += __shfl_xor(sw, 1);
            scur = scur * scale + sw;
            mcur = mnew;

            // y[128] striped: lane holds cols 4*lane..4*lane+3
            y.x *= scale; y.y *= scale; y.z *= scale; y.w *= scale;
            const float* col = buf + lane * 4;
#pragma unroll
            for (int m = 0; m < 16; ++m) {
                float  wm = __shfl(w, m);
                float4 xm = *(const float4*)(col + m * HIDDEN);
                y.x += wm * xm.x; y.y += wm * xm.y;
                y.z += wm * xm.z; y.w += wm * xm.w;
            }
        }
    }
    // partial record: [m, s, pad, pad, y[128]]  (132 floats)
    float* rec = partials + (size_t)gw * 132;
    if (lane == 0) { rec[0] = mcur; rec[1] = scur; }
    *(float4*)(rec + 4 + lane * 4) = y;
}

// ---------------------------------------------------------------------------
// Kernel 2: deterministic fixed-order reduction of P wave partials, then
// h = (Y/S) @ Wv, out = h @ Wo.
// ---------------------------------------------------------------------------
__global__ __launch_bounds__(128) void finish_kernel(
    const float* __restrict__ part, const float* __restrict__ Wv,
    const float* __restrict__ Wo, float* __restrict__ out, int P)
{
    __shared__ float e[2048];
    __shared__ float red[128];
    __shared__ float shy[128];
    __shared__ float shh[128];
    const int t = threadIdx.x;

    float mloc = -INFINITY;
    for (int p = t; p < P; p += 128) mloc = fmaxf(mloc, part[(size_t)p * 132]);
    red[t] = mloc; __syncthreads();
    for (int s = 64; s > 0; s >>= 1) { if (t < s) red[t] = fmaxf(red[t], red[t + s]); __syncthreads(); }
    const float M = red[0]; __syncthreads();

    float sloc = 0.f;
    for (int p = t; p < P; p += 128) {
        float ep = __expf(part[(size_t)p * 132] - M);
        e[p] = ep;
        sloc += part[(size_t)p * 132 + 1] * ep;
    }
    red[t] = sloc; __syncthreads();
    for (int s = 64; s > 0; s >>= 1) { if (t < s) red[t] += red[t + s]; __syncthreads(); }
    const float S = red[0]; __syncthreads();

    float Y = 0.f;
    for (int p = 0; p < P; ++p) Y += e[p] * part[(size_t)p * 132 + 4 + t];
    shy[t] = Y / S; __syncthreads();

    float h = 0.f;
    for (int j = 0; j < HIDDEN; ++j) h += shy[j] * Wv[j * HIDDEN + t];
    shh[t] = h; __syncthreads();

    float o = 0.f;
    for (int n = 0; n < HIDDEN; ++n) o += shh[n] * Wo[n * HIDDEN + t];
    out[t] = o;
}

// ---------------------------------------------------------------------------
extern "C" void kernel_launch(void* const* d_in, const int* in_sizes, int n_in,
                              void* d_out, int out_size, void* d_ws, size_t ws_size,
                              hipStream_t stream)
{
    const float* X  = (const float*)d_in[0];
    const float* xv = (const float*)d_in[1];
    const float* Wq = (const float*)d_in[2];
    const float* Wk = (const float*)d_in[3];
    const float* Wv = (const float*)d_in[4];
    const float* Wo = (const float*)d_in[5];
    const int* visited = (const int*)d_in[6];
    const int* snp = (const int*)d_in[7];
    const int* pnp = (const int*)d_in[8];
    float* outp = (float*)d_out;

    const int nodes    = in_sizes[0] / HIDDEN;   // 200000 (multiple of 16)
    const int nVisited = in_sizes[6];
    const int tiles    = nodes / 16;             // 12500
    const int nWords   = (nodes + 31) / 32;      // 6250

    // workspace layout: qk(512B) | bitmask(nWords, padded to 128B) | partials
    float*    qk    = (float*)d_ws;
    unsigned* maskW = (unsigned*)((char*)d_ws + 512);
    size_t maskBytes = (((size_t)nWords * 4) + 127) & ~(size_t)127;
    float* partials  = (float*)((char*)d_ws + 512 + maskBytes);

    int P = 2048;                                // total waves (deterministic)
    size_t head = 512 + maskBytes;
    if (ws_size > head) {
        size_t maxP = (ws_size - head) / (132 * 4);
        if ((size_t)P > maxP) P = (int)maxP;
    }
    P &= ~1; if (P < 2) P = 2;
    const int span = (tiles + P - 1) / P;

    prep_kernel<<<1, 256, 0, stream>>>(X, xv, Wq, Wk, visited, snp, pnp,
                                       qk, maskW, nWords, nVisited);
    // 2 waves/block, 2x8KB double buffer per wave + qk = 33280 B LDS
    stream_kernel<<<P / 2, 64, 33280, stream>>>(X, qk, maskW, partials, span, tiles);
    finish_kernel<<<1, 128, 0, stream>>>(partials, Wv, Wo, outp, P);
}